// ECLay_69406671503388
// MI455X (gfx1250) — compile-verified
//
#include <hip/hip_runtime.h>
#include <hip/hip_bf16.h>

// ---------------------------------------------------------------------------
// ECLay for MI455X (gfx1250):
//   Pass 1: Euler-characteristic histogram (V - E + F per threshold bin).
//           Column-walking threads: top/top-left neighbors carried in
//           registers, left neighbor via wave32 shuffle -> ~1 coalesced
//           global load per pixel. Per-thread private LDS histograms with
//           bin-cancellation (~2 ds_add atomics per pixel expected).
//   Pass 2: single block -- cumsum -> EC curves (32x96 fp32), then the MLP
//           as two fp32 WMMA GEMMs (v_wmma_f32_16x16x4_f32), bias+ReLU fused.
// ---------------------------------------------------------------------------

typedef __attribute__((ext_vector_type(2))) float v2f;
typedef __attribute__((ext_vector_type(8))) float v8f;

#define HH 512
#define WW 512
#define NT 32            // thresholds
#define NBC 96           // B*C
#define HSTRIDE 33       // 32 bins + 1 trash slot (absorbs inf / rounding edge)

// bin of value v under thresholds linspace(0,7,32); 32 == trash slot
__device__ __forceinline__ int binOf(float v) {
    if (!(v < 7.0f)) return 32;
    int k = (int)(v * (31.0f / 7.0f)) + 1;   // first t_k with t_k > v
    return k > 32 ? 32 : k;
}

__global__ void __launch_bounds__(256) ec_zero(int* g, int n) {
    int i = blockIdx.x * 256 + threadIdx.x;
    if (i < n) g[i] = 0;
}

// grid (8, 96): blockIdx.y = image (b*C+c), blockIdx.x = 64-row slab.
// 512 threads, thread tid owns column j = tid and walks 64 rows.
__global__ void __launch_bounds__(512) ec_hist(const float* __restrict__ x,
                                               int* __restrict__ ghist) {
    __shared__ int lh[512 * HSTRIDE];            // per-thread private histograms
    const int tid  = threadIdx.x;
    const int lane = tid & 31;
    int* myh = &lh[tid * HSTRIDE];
#pragma unroll
    for (int k = 0; k < HSTRIDE; ++k) myh[k] = 0;
    __syncthreads();

    const int bc   = blockIdx.y;
    const int row0 = blockIdx.x * 64;
    const int j    = tid;                        // column
    const float* px = x + (size_t)bc * (HH * WW);
    const float INF = 3.4e38f;

    // prologue: top / top-left neighbors for the first row of the slab
    float tp, tl;
    if (row0 == 0) {                             // uniform per block
        tp = INF; tl = INF;
    } else {
        tp = px[(row0 - 1) * WW + j];
        float t = __shfl_up(tp, 1, 32);
        if (lane == 0) t = (j > 0) ? px[(row0 - 1) * WW + j - 1] : INF;
        tl = t;
    }

    for (int it = 0; it < 64; ++it) {
        const int i = row0 + it;
        float c = px[i * WW + j];                // one coalesced load / pixel
        float l = __shfl_up(c, 1, 32);           // left neighbor from lane-1
        if (lane == 0) l = (j > 0) ? px[i * WW + j - 1] : INF;

        float ew = fminf(l, c);                  // left edge value
        float eh = fminf(tp, c);                 // top edge value
        float vv = fminf(fminf(tl, tp), ew);     // top-left vertex value

        int kc = binOf(c), ke = binOf(ew), kh = binOf(eh), kv = binOf(vv);

        // (+1 face @ kc, -1 edge @ ke) and (+1 vertex @ kv, -1 edge @ kh):
        // ordered pairs cancel whenever they land in the same bin.
        if (kc != ke) { atomicAdd(&myh[kc], 1); atomicAdd(&myh[ke], -1); }
        if (kv != kh) { atomicAdd(&myh[kv], 1); atomicAdd(&myh[kh], -1); }

        // boundary cells owned by border pixels (same cancellation trick):
        if (j == WW - 1 && kc != kh) {           // e_w(i,W) @ c, vert(i,W) @ eh
            atomicAdd(&myh[kc], -1); atomicAdd(&myh[kh], 1);
        }
        if (i == HH - 1) {                       // e_h(H,j) @ c, vert(H,j) @ ew
            if (kc != ke) { atomicAdd(&myh[kc], -1); atomicAdd(&myh[ke], 1); }
            if (j == WW - 1) atomicAdd(&myh[kc], 1);   // vert(H,W) @ c
        }

        tp = c; tl = l;                          // carry to next row
    }
    __syncthreads();

    // reduce 512 private columns -> 32 bins (16 partials/bin) -> global atomics
    int bin = tid & 31, grp = tid >> 5;
    int s = 0;
#pragma unroll
    for (int q = 0; q < 32; ++q) s += lh[(grp * 32 + q) * HSTRIDE + bin];
    if (s) atomicAdd(&ghist[bc * NT + bin], s);
}

// one block, 256 threads: cumsum -> EC matrix, then MLP via fp32 WMMA
__global__ void __launch_bounds__(256) ec_mlp(const int* __restrict__ ghist,
                                              const float* __restrict__ W1,
                                              const float* __restrict__ b1,
                                              const float* __restrict__ W2,
                                              const float* __restrict__ b2,
                                              float* __restrict__ out) {
    __shared__ float ec[32 * 96];   // A-matrix of GEMM1: row b, col c*32+t
    __shared__ float h1[32 * 64];   // hidden activations
    const int tid = threadIdx.x;

    if (tid < NBC) {                // 96 threads: prefix-sum the histogram
        int b = tid / 3, c = tid % 3;
        int s = 0;
#pragma unroll
        for (int t = 0; t < NT; ++t) {
            s += ghist[tid * NT + t];
            ec[b * 96 + c * NT + t] = (float)s;
        }
    }
    __syncthreads();

    const int wave = tid >> 5, lane = tid & 31;
    const int l16 = lane & 15, kh = lane >> 4;   // A frag: lanes 0-15 K={0,1}, 16-31 K={2,3}

    // ---- GEMM1: (32x96) @ W1^T (96x64), 8 waves = 2x4 tiles of 16x16 ----
    {
        const int m0 = (wave >> 2) * 16, n0 = (wave & 3) * 16;
        v8f acc = {};
        for (int k0 = 0; k0 < 96; k0 += 4) {
            int ka = k0 + 2 * kh;
            v2f a, bf;
            a.x  = ec[(m0 + l16) * 96 + ka];
            a.y  = ec[(m0 + l16) * 96 + ka + 1];
            bf.x = W1[(n0 + l16) * 96 + ka];     // Bmat[k][n] = W1[n][k]
            bf.y = W1[(n0 + l16) * 96 + ka + 1];
            acc = __builtin_amdgcn_wmma_f32_16x16x4_f32(
                false, a, false, bf, (short)0, acc, false, false);
        }
        float bias = b1[n0 + l16];
#pragma unroll
        for (int r = 0; r < 8; ++r) {            // D: VGPR r -> rows r / r+8
            int m = m0 + r + 8 * kh;
            float v = acc[r] + bias;
            h1[m * 64 + (n0 + l16)] = v > 0.f ? v : 0.f;
        }
    }
    __syncthreads();

    // ---- GEMM2: (32x64) @ W2^T (64x32), 4 waves = 2x2 tiles ----
    if (wave < 4) {
        const int m0 = (wave >> 1) * 16, n0 = (wave & 1) * 16;
        v8f acc = {};
        for (int k0 = 0; k0 < 64; k0 += 4) {
            int ka = k0 + 2 * kh;
            v2f a, bf;
            a.x  = h1[(m0 + l16) * 64 + ka];
            a.y  = h1[(m0 + l16) * 64 + ka + 1];
            bf.x = W2[(n0 + l16) * 64 + ka];     // Bmat[k][m] = W2[m][k]
            bf.y = W2[(n0 + l16) * 64 + ka + 1];
            acc = __builtin_amdgcn_wmma_f32_16x16x4_f32(
                false, a, false, bf, (short)0, acc, false, false);
        }
        float bias = b2[n0 + l16];
#pragma unroll
        for (int r = 0; r < 8; ++r) {
            int m = m0 + r + 8 * kh;
            out[m * 32 + (n0 + l16)] = acc[r] + bias;
        }
    }
}

extern "C" void kernel_launch(void* const* d_in, const int* in_sizes, int n_in,
                              void* d_out, int out_size, void* d_ws, size_t ws_size,
                              hipStream_t stream) {
    const float* x  = (const float*)d_in[0];   // (32,3,512,512)
    // d_in[1] = tseq: linspace(0,7,32) -> folded into analytic binning
    const float* W1 = (const float*)d_in[2];   // (64,96)
    const float* b1 = (const float*)d_in[3];   // (64)
    const float* W2 = (const float*)d_in[4];   // (32,64)
    const float* b2 = (const float*)d_in[5];   // (32)
    float* out = (float*)d_out;                // (32,32)

    int* ghist = (int*)d_ws;                   // 96*32 ints

    ec_zero<<<dim3((NBC * NT + 255) / 256), 256, 0, stream>>>(ghist, NBC * NT);
    ec_hist<<<dim3(8, NBC), 512, 0, stream>>>(x, ghist);
    ec_mlp<<<dim3(1), 256, 0, stream>>>(ghist, W1, b1, W2, b2, out);
}